// SpGraphAttentionLayer_42434276884994
// MI455X (gfx1250) — compile-verified
//
#include <hip/hip_runtime.h>

// Problem constants (match reference)
#define NN      100000
#define EE      1600000
#define DD      128
#define NEG_SLOPE 0.2f

typedef float v2f __attribute__((ext_vector_type(2)));
typedef float v8f __attribute__((ext_vector_type(8)));

// ---------------------------------------------------------------------------
// h = x @ W  via V_WMMA_F32_16X16X4_F32.
// One wave computes one 16x16 tile of h; 8 waves per block -> one 16-row strip
// covers all 8 column tiles. K-loop: 32 steps of K=4.
//
// A (16x4 f32) VGPR layout (ISA 7.12.2): lanes 0-15 = M rows, VGPR0/1 hold
// K = 2*(lane>>4) and K+1  -> contiguous float2 load per lane.
// B (4x16): VGPR0 lanes0-15 = row K=2*(lane>>4), VGPR1 = row K+1, cols = lane&15.
// C/D (16x16 f32): VGPR r -> M = r + 8*(lane>>4), N = lane&15.
// ---------------------------------------------------------------------------
__global__ void gat_gemm_wmma(const float* __restrict__ x,
                              const float* __restrict__ W,
                              float* __restrict__ h) {
    const int lane = threadIdx.x & 31;
    const int wave = threadIdx.x >> 5;
    const int row0 = blockIdx.x << 4;   // 6250 row strips
    const int col0 = wave << 4;         // 8 col tiles
    const int r    = lane & 15;
    const int hi   = lane >> 4;

    v8f acc = {};
    const float* arow = x + (size_t)(row0 + r) * DD;

#pragma unroll 4
    for (int k = 0; k < DD; k += 4) {
        const int ka = k + 2 * hi;
        // A fragment: two consecutive K values for row r
        const float2 af = *(const float2*)(arow + ka);
        v2f a; a.x = af.x; a.y = af.y;
        // B fragment: rows ka and ka+1 of W, column col0 + r
        v2f b;
        b.x = W[(size_t)ka * DD + col0 + r];
        b.y = W[(size_t)(ka + 1) * DD + col0 + r];
        acc = __builtin_amdgcn_wmma_f32_16x16x4_f32(
            /*neg_a=*/false, a, /*neg_b=*/false, b,
            /*c_mod=*/(short)0, acc, /*reuse_a=*/false, /*reuse_b=*/false);
    }

    // Store D tile
    float* hout = h + (size_t)(row0 + 8 * hi) * DD + col0 + r;
#pragma unroll
    for (int i = 0; i < 8; ++i)
        hout[(size_t)i * DD] = acc[i];
}

// ---------------------------------------------------------------------------
// s[row] = dot(h[row], a1), t[row] = dot(h[row], a2). One wave per row.
// ---------------------------------------------------------------------------
__global__ void gat_st(const float* __restrict__ h,
                       const float* __restrict__ a,
                       float* __restrict__ s,
                       float* __restrict__ t) {
    const int lane = threadIdx.x & 31;
    const int wave = threadIdx.x >> 5;
    const int row  = blockIdx.x * 8 + wave;   // 12500 blocks * 8 waves = 100000

    const float4 hv  = *(const float4*)(h + (size_t)row * DD + lane * 4);
    const float4 a1v = *(const float4*)(a + lane * 4);
    const float4 a2v = *(const float4*)(a + DD + lane * 4);

    float ps = hv.x * a1v.x + hv.y * a1v.y + hv.z * a1v.z + hv.w * a1v.w;
    float pt = hv.x * a2v.x + hv.y * a2v.y + hv.z * a2v.z + hv.w * a2v.w;
#pragma unroll
    for (int off = 16; off > 0; off >>= 1) {
        ps += __shfl_xor(ps, off, 32);
        pt += __shfl_xor(pt, off, 32);
    }
    if (lane == 0) { s[row] = ps; t[row] = pt; }
}

// ---------------------------------------------------------------------------
// edge_e[i] = exp(-leaky_relu(s[src] + t[dst]))
// ---------------------------------------------------------------------------
__global__ void gat_edge(const int* __restrict__ edge,
                         const float* __restrict__ s,
                         const float* __restrict__ t,
                         float* __restrict__ edge_e) {
    const int i = blockIdx.x * blockDim.x + threadIdx.x;   // exactly EE threads
    const int src = edge[i];
    const int dst = edge[EE + i];
    const float sc = s[src] + t[dst];
    const float lr = sc > 0.0f ? sc : NEG_SLOPE * sc;
    edge_e[i] = __expf(-lr);
}

// ---------------------------------------------------------------------------
// h_prime[src] += edge_e * h[dst]. One wave per edge; each lane handles 4
// floats -> 128 fp32 atomics per edge into L2-resident h_prime.
// ---------------------------------------------------------------------------
__global__ void gat_aggregate(const int* __restrict__ edge,
                              const float* __restrict__ h,
                              const float* __restrict__ edge_e,
                              float* __restrict__ hp) {
    const int lane = threadIdx.x & 31;
    const int wave = threadIdx.x >> 5;
    const int e    = blockIdx.x * 8 + wave;   // 200000 blocks * 8 = EE

    const int src = edge[e];
    const int dst = edge[EE + e];
    const float w = edge_e[e];

    const float4 hv = *(const float4*)(h + (size_t)dst * DD + lane * 4);
    float* base = hp + (size_t)src * DD + lane * 4;
    atomicAdd(base + 0, w * hv.x);
    atomicAdd(base + 1, w * hv.y);
    atomicAdd(base + 2, w * hv.z);
    atomicAdd(base + 3, w * hv.w);
}

// ---------------------------------------------------------------------------
// Zero the h_prime accumulator region (float4 granularity).
// ---------------------------------------------------------------------------
__global__ void gat_zero(float4* __restrict__ p) {
    p[(size_t)blockIdx.x * blockDim.x + threadIdx.x] = float4{0.f, 0.f, 0.f, 0.f};
}

// ---------------------------------------------------------------------------
// In-place ELU (alpha = 1) on h_prime.
// ---------------------------------------------------------------------------
__global__ void gat_elu(float4* __restrict__ p) {
    const size_t i = (size_t)blockIdx.x * blockDim.x + threadIdx.x;
    float4 v = p[i];
    v.x = v.x > 0.f ? v.x : __expf(v.x) - 1.f;
    v.y = v.y > 0.f ? v.y : __expf(v.y) - 1.f;
    v.z = v.z > 0.f ? v.z : __expf(v.z) - 1.f;
    v.w = v.w > 0.f ? v.w : __expf(v.w) - 1.f;
    p[i] = v;
}

extern "C" void kernel_launch(void* const* d_in, const int* in_sizes, int n_in,
                              void* d_out, int out_size, void* d_ws, size_t ws_size,
                              hipStream_t stream) {
    // setup_inputs order: edge (int), x (f32), W (f32), a (f32)
    const int*   edge = (const int*)d_in[0];
    const float* x    = (const float*)d_in[1];
    const float* W    = (const float*)d_in[2];
    const float* a    = (const float*)d_in[3];

    // Workspace layout: h [N*D], s [N], t [N]  (~52 MB)
    float* h = (float*)d_ws;
    float* s = h + (size_t)NN * DD;
    float* t = s + NN;

    // Output layout (tuple, flattened): elu(h_prime) [N*D] then edge_e [E]
    float* hp     = (float*)d_out;
    float* edge_e = hp + (size_t)NN * DD;

    // 1) zero h_prime accumulator: N*D/4 float4 = 3.2M -> 12500 blocks
    gat_zero<<<(NN * DD / 4) / 256, 256, 0, stream>>>((float4*)hp);

    // 2) h = x @ W : 6250 row strips x 8 waves/block
    gat_gemm_wmma<<<NN / 16, 256, 0, stream>>>(x, W, h);

    // 3) s, t GEMVs: 8 rows per block
    gat_st<<<NN / 8, 256, 0, stream>>>(h, a, s, t);

    // 4) per-edge attention coefficients (written directly into d_out tail)
    gat_edge<<<EE / 256, 256, 0, stream>>>(edge, s, t, edge_e);

    // 5) scatter-add aggregation: one wave per edge
    gat_aggregate<<<EE / 8, 256, 0, stream>>>(edge, h, edge_e, hp);

    // 6) in-place ELU on h_prime
    gat_elu<<<(NN * DD / 4) / 256, 256, 0, stream>>>((float4*)hp);
}